// BilateralSolverLocal_15204184227844
// MI455X (gfx1250) — compile-verified
//
#include <hip/hip_runtime.h>

// ---------------------------------------------------------------------------
// Bilateral smoothness + fidelity loss, MI455X (gfx1250, wave32).
// Memory-bound: streams 426 MB of w_ij once (NT b128 loads), output image
// (1 MB) stays hot in L2. Reductions use V_WMMA_F32_16X16X4_F32 with an
// all-ones B matrix as an exact f32 32-lane summation (matrix pipe), then a
// deterministic two-pass partial-sum scheme (no float atomics).
// ---------------------------------------------------------------------------

typedef float v2f __attribute__((ext_vector_type(2)));
typedef float v4f __attribute__((ext_vector_type(4)));
typedef float v8f __attribute__((ext_vector_type(8)));

#define HH      512
#define WWD     512
#define KWIN    21
#define COFF    10           // (K-1)/2
#define HO      492
#define WO      492
#define NCH     440          // K*K - 1
#define PLANE   (HO * WO)    // 242064
#define Q4      (PLANE / 4)  // 60516 float4s per channel plane
#define ROWQ    (WO / 4)     // 123 float4s per row
#define CTR_T   (COFF * KWIN + COFF)   // 220

#define THREADS   256
#define BLOCKS_X  32
#define SM_PART   (NCH * BLOCKS_X)     // 14080 smooth partials
#define FID_BLK   64                   // fidelity partial blocks

// ---- exact f32 sum of all 32 lanes via the matrix pipe ---------------------
// A (16x4 f32): lane L%16 rows; our accumulator in A[:,0]/A[:,2], zeros in
// A[:,1]/A[:,3]. B = ones (4x16). D[m,n] = sum_k A[m,k]. Each lane then sums
// its 8 D VGPRs (rows m in its half, fixed column), and lane L + lane L^16
// yields the full 32-lane total (same value in every lane).
__device__ __forceinline__ float wave_sum32_wmma(float x) {
    v2f a; a[0] = x;    a[1] = 0.0f;
    v2f b; b[0] = 1.0f; b[1] = 1.0f;
    v8f c = {};
    c = __builtin_amdgcn_wmma_f32_16x16x4_f32(
            /*neg_a=*/false, a, /*neg_b=*/false, b,
            /*c_mod=*/(short)0, c, /*reuse_a=*/false, /*reuse_b=*/false);
    float s = ((c[0] + c[1]) + (c[2] + c[3])) + ((c[4] + c[5]) + (c[6] + c[7]));
    s += __shfl_xor(s, 16, 32);
    return s;
}

// Deterministic block sum (valid in thread 0). blockDim.x == THREADS.
__device__ __forceinline__ float block_sum(float x) {
    __shared__ float wsum[THREADS / 32];
    float w = wave_sum32_wmma(x);          // EXEC all-ones here (reconverged)
    const int lane = threadIdx.x & 31;
    const int wid  = threadIdx.x >> 5;
    if (lane == 0) wsum[wid] = w;
    __syncthreads();
    float r = 0.0f;
    if (threadIdx.x == 0) {
        #pragma unroll
        for (int i = 0; i < THREADS / 32; ++i) r += wsum[i];
    }
    return r;
}

// ---- kernel 1: smooth-term partial sums, one tap channel per blockIdx.y ----
__global__ void __launch_bounds__(THREADS)
smooth_partials_kernel(const float* __restrict__ img,
                       const float* __restrict__ wij,
                       const float* __restrict__ bias,
                       float* __restrict__ partials) {
    const int ch = blockIdx.y;                       // 0..439
    const int t  = ch + (ch >= CTR_T ? 1 : 0);       // skip center tap
    const int dy = t / KWIN;
    const int dx = t % KWIN;
    const float bs = bias[ch];
    const float* __restrict__ wch = wij + (size_t)ch * PLANE;
    const bool nb_even = ((dx & 1) == 0);            // wave-uniform

    const int stride = BLOCKS_X * THREADS;           // 8192
    float acc = 0.0f;

    for (int q = blockIdx.x * THREADS + threadIdx.x; q < Q4; q += stride) {
        // stream w_ij non-temporally (touched exactly once, 426 MB total)
        v4f w4 = __builtin_nontemporal_load((const v4f*)(wch + 4 * q));
        if (q + stride < Q4)
            __builtin_prefetch(wch + 4 * (q + stride), 0, 0);

        const int y = q / ROWQ;
        const int x = (q - y * ROWQ) * 4;            // multiple of 4

        // center pixels: offset (y+10)*512 + (x+10) is even -> 8B-aligned
        const float* ctr = img + (y + COFF) * WWD + (x + COFF);
        v2f c01 = *(const v2f*)(ctr);
        v2f c23 = *(const v2f*)(ctr + 2);

        const float* nb = img + (y + dy) * WWD + (x + dx);
        float n0, n1, n2, n3;
        if (nb_even) {                               // even dx -> 8B-aligned
            v2f a0 = *(const v2f*)(nb);
            v2f a1 = *(const v2f*)(nb + 2);
            n0 = a0[0]; n1 = a0[1]; n2 = a1[0]; n3 = a1[1];
        } else {
            n0 = nb[0]; n1 = nb[1]; n2 = nb[2]; n3 = nb[3];
        }

        const float d0 = (c01[0] - n0) + bs;
        const float d1 = (c01[1] - n1) + bs;
        const float d2 = (c23[0] - n2) + bs;
        const float d3 = (c23[1] - n3) + bs;
        acc = fmaf(w4[0] * d0, d0, acc);
        acc = fmaf(w4[1] * d1, d1, acc);
        acc = fmaf(w4[2] * d2, d2, acc);
        acc = fmaf(w4[3] * d3, d3, acc);
    }

    const float r = block_sum(acc);
    if (threadIdx.x == 0)
        partials[blockIdx.y * gridDim.x + blockIdx.x] = r;
}

// ---- kernel 2: fidelity-term partial sums ----------------------------------
__global__ void __launch_bounds__(THREADS)
fid_partials_kernel(const float* __restrict__ img,
                    const float* __restrict__ tgt,
                    float* __restrict__ partials) {
    const int N4 = (HH * WWD) / 4;                   // 65536
    float acc = 0.0f;
    for (int q = blockIdx.x * THREADS + threadIdx.x; q < N4;
         q += FID_BLK * THREADS) {
        v4f o = *(const v4f*)(img + 4 * q);
        v4f t = *(const v4f*)(tgt + 4 * q);
        #pragma unroll
        for (int i = 0; i < 4; ++i) {
            const float d = o[i] - t[i];
            acc = fmaf(d, d, acc);
        }
    }
    const float r = block_sum(acc);
    if (threadIdx.x == 0) partials[SM_PART + blockIdx.x] = r;
}

// ---- kernel 3: deterministic final fold to the scalar loss -----------------
__global__ void __launch_bounds__(THREADS)
final_reduce_kernel(const float* __restrict__ partials,
                    float* __restrict__ out) {
    // smooth = (H*W*LAM / (NCH*PLANE)) * sum(w d^2);  fidelity = sum/(H*W)
    constexpr double SM_SCALE_D =
        ((double)HH * (double)WWD * 128.0) / ((double)NCH * (double)PLANE);
    const float SM_SCALE  = (float)SM_SCALE_D;
    const float FID_SCALE = 1.0f / (float)(HH * WWD);

    float acc_s = 0.0f, acc_f = 0.0f;
    for (int i = threadIdx.x; i < SM_PART; i += THREADS) acc_s += partials[i];
    for (int i = threadIdx.x; i < FID_BLK; i += THREADS)
        acc_f += partials[SM_PART + i];

    const float v = SM_SCALE * acc_s + FID_SCALE * acc_f;
    const float r = block_sum(v);
    if (threadIdx.x == 0) out[0] = r;
}

// ---------------------------------------------------------------------------
extern "C" void kernel_launch(void* const* d_in, const int* in_sizes, int n_in,
                              void* d_out, int out_size, void* d_ws, size_t ws_size,
                              hipStream_t stream) {
    const float* img  = (const float*)d_in[0];   // output (H,W)
    const float* tgt  = (const float*)d_in[1];   // target (H,W)
    const float* wij  = (const float*)d_in[2];   // (440, 492, 492)
    const float* bias = (const float*)d_in[3];   // (440,)
    float* partials   = (float*)d_ws;            // 14080 + 64 floats
    float* out        = (float*)d_out;           // scalar

    dim3 g1(BLOCKS_X, NCH);
    smooth_partials_kernel<<<g1, THREADS, 0, stream>>>(img, wij, bias, partials);
    fid_partials_kernel<<<FID_BLK, THREADS, 0, stream>>>(img, tgt, partials);
    final_reduce_kernel<<<1, THREADS, 0, stream>>>(partials, out);
}